// ClassConditionalBatchNorm2d_76192719831914
// MI455X (gfx1250) — compile-verified
//
#include <hip/hip_runtime.h>
#include <stdint.h>

typedef float v4f __attribute__((ext_vector_type(4)));
typedef float v2f __attribute__((ext_vector_type(2)));

#define CCBN_EPS 1e-5f
#define CCBN_MIN_COUNT 100.0f

// ---------------------------------------------------------------------------
// Kernel A: fold all stats logic into per-(b,c) {scale, shift}.
// Grid: (B, ceil(C/256)) x 256 threads -> no integer division anywhere.
//   scale = weight[c] / sqrt(var + eps)
//   shift = bias[c] - mean * scale
// ---------------------------------------------------------------------------
__global__ void ccbn_prep(const int* __restrict__ labels,
                          const float* __restrict__ weight,
                          const float* __restrict__ bias,
                          const float* __restrict__ gmean,
                          const float* __restrict__ gvar,
                          const float* __restrict__ cmean,
                          const float* __restrict__ cvar,
                          const float* __restrict__ counts,
                          const float* __restrict__ alpha_p,
                          v2f* __restrict__ ss,
                          int C) {
    const int b = blockIdx.x;
    const int c = blockIdx.y * blockDim.x + threadIdx.x;
    if (c >= C) return;

    const float alpha = alpha_p[0];
    const int   lab   = labels[b];
    const float cnt   = counts[lab];

    float m, v;
    if ((alpha > 0.0f) && (cnt >= CCBN_MIN_COUNT)) {
        const float om = 1.0f - alpha;
        const size_t o = (size_t)lab * (size_t)C + (size_t)c;
        m = om * gmean[c] + alpha * cmean[o];
        v = om * gvar[c]  + alpha * cvar[o];
        v = fmaxf(v, CCBN_EPS);           // jnp.clip(..., EPS, None)
    } else {
        m = gmean[c];
        v = gvar[c];
    }
    const float inv = 1.0f / sqrtf(v + CCBN_EPS);
    const float sc  = weight[c] * inv;
    v2f r;
    r.x = sc;
    r.y = fmaf(-m, sc, bias[c]);
    ss[(size_t)b * (size_t)C + (size_t)c] = r;
}

// ---------------------------------------------------------------------------
// Kernel B: streaming apply. 4-buffer ring, 3 outstanding async copies per
// wave, ONE barrier per 4KB chunk. All chunks are full (uniform EXEC), so
// ASYNCcnt is uniform across the waves of a block.
//
// Iteration s:
//   s_wait_asynccnt <=2        -> chunk s has landed in LDS (loads retire
//                                 in order; 2 newer may be outstanding)
//   __syncthreads()            -> every wave's slice of chunk s is visible;
//                                 also: every wave finished READING buffer
//                                 (s-1)&3 before signaling (its fma forced
//                                 the dscnt wait), so that buffer is free
//   issue chunk s+3 -> buf[(s+3)&3] == buf[(s-1)&3]
//   consume chunk s from buf[s&3], fma, NT b128 store
// ---------------------------------------------------------------------------
#define CCBN_CHUNK 256   // float4 per chunk == blockDim.x  (4 KB)
#define CCBN_KCH   16    // chunks per block (16 KB LDS ring)

__device__ __forceinline__ void ccbn_async_b128(const v4f* gsrc, v4f* lds_dst) {
    uint32_t lds = (uint32_t)(uintptr_t)lds_dst;          // low 32 bits = LDS offset
    uint64_t ga  = (uint64_t)(uintptr_t)gsrc;
    asm volatile("global_load_async_to_lds_b128 %0, %1, off th:TH_LOAD_NT"
                 :: "v"(lds), "v"(ga) : "memory");
}

__global__ __launch_bounds__(CCBN_CHUNK) void ccbn_apply_async(
    const v4f* __restrict__ x4, const v2f* __restrict__ ss,
    v4f* __restrict__ o4, uint32_t magicM, uint32_t magicS) {
    __shared__ v4f buf[4][CCBN_CHUNK];

    const uint32_t tid  = threadIdx.x;
    const uint32_t base = (uint32_t)blockIdx.x * (uint32_t)CCBN_KCH;

    // Prime the pipeline: chunks 0,1,2 -> buffers 0,1,2.
    ccbn_async_b128(x4 + (size_t)((base + 0u) * CCBN_CHUNK + tid), &buf[0][tid]);
    ccbn_async_b128(x4 + (size_t)((base + 1u) * CCBN_CHUNK + tid), &buf[1][tid]);
    ccbn_async_b128(x4 + (size_t)((base + 2u) * CCBN_CHUNK + tid), &buf[2][tid]);

#pragma unroll
    for (int s = 0; s < CCBN_KCH; ++s) {
        if (s < CCBN_KCH - 2)       asm volatile("s_wait_asynccnt 0x2" ::: "memory");
        else if (s == CCBN_KCH - 2) asm volatile("s_wait_asynccnt 0x1" ::: "memory");
        else                        asm volatile("s_wait_asynccnt 0x0" ::: "memory");

        __syncthreads();

        if (s + 3 < CCBN_KCH) {
            ccbn_async_b128(x4 + (size_t)((base + (uint32_t)s + 3u) * CCBN_CHUNK + tid),
                            &buf[(s + 3) & 3][tid]);
        }

        const uint32_t g = (base + (uint32_t)s) * CCBN_CHUNK + tid;
        const v4f val = buf[s & 3][tid];                       // ds_load_b128
        const uint32_t plane = (uint32_t)(((uint64_t)g * magicM) >> magicS); // g / hw4
        const v2f w = ss[plane];
        v4f r;
        r.x = fmaf(val.x, w.x, w.y);
        r.y = fmaf(val.y, w.x, w.y);
        r.z = fmaf(val.z, w.x, w.y);
        r.w = fmaf(val.w, w.x, w.y);
        __builtin_nontemporal_store(r, o4 + (size_t)g);        // NT b128 store
    }
}

// Generic scalar tail (remainder elements); not launched for harness shapes.
__global__ void ccbn_tail(const float* __restrict__ x, const v2f* __restrict__ ss,
                          float* __restrict__ o, uint32_t start, uint32_t n,
                          uint32_t hw) {
    uint32_t i = start + blockIdx.x * blockDim.x + threadIdx.x;
    if (i >= n) return;
    v2f w = ss[i / hw];
    o[i] = fmaf(x[i], w.x, w.y);
}

// ---------------------------------------------------------------------------
// Host: magic constant for unsigned division by d, valid for all g <= nmax.
// ---------------------------------------------------------------------------
static void ccbn_magic_u32(uint32_t d, uint64_t nmax, uint32_t* M, uint32_t* sh) {
    *M = 0; *sh = 0;
    if (d == 0) return;
    for (uint32_t s = 32; s < 63; ++s) {
        unsigned __int128 pow2 = ((unsigned __int128)1) << s;
        uint64_t m = (uint64_t)((pow2 + d - 1) / d);           // ceil(2^s / d)
        if (m >> 32) continue;                                  // must fit u32
        unsigned __int128 e = (unsigned __int128)m * d - pow2;  // error term
        if (e * nmax < pow2) { *M = (uint32_t)m; *sh = s; return; }
    }
}

extern "C" void kernel_launch(void* const* d_in, const int* in_sizes, int n_in,
                              void* d_out, int out_size, void* d_ws, size_t ws_size,
                              hipStream_t stream) {
    const float* x      = (const float*)d_in[0];
    const int*   labels = (const int*)  d_in[1];
    const float* weight = (const float*)d_in[2];
    const float* bias   = (const float*)d_in[3];
    const float* gmean  = (const float*)d_in[4];
    const float* gvar   = (const float*)d_in[5];
    const float* cmean  = (const float*)d_in[6];
    const float* cvar   = (const float*)d_in[7];
    const float* counts = (const float*)d_in[8];
    const float* alpha  = (const float*)d_in[9];

    const int B = in_sizes[1];          // labels: [B]
    const int C = in_sizes[2];          // weight: [C]
    const uint32_t N  = (uint32_t)out_size;
    const uint32_t HW = N / (uint32_t)(B * C);

    v2f* ss = (v2f*)d_ws;               // B*C float2 = 128 KB for harness shapes

    // Kernel A: per-(b,c) scale/shift (no divisions: 2D grid mapping)
    {
        dim3 grid((unsigned)B, (unsigned)((C + 255) / 256));
        ccbn_prep<<<grid, 256, 0, stream>>>(
            labels, weight, bias, gmean, gvar, cmean, cvar, counts, alpha, ss, C);
    }

    // Kernel B: async-pipelined stream over float4s
    uint32_t covered4 = 0;
    if ((HW % 4u) == 0u) {
        const uint32_t N4     = N / 4u;
        const uint32_t hw4    = HW / 4u;                 // 784 for harness
        const uint32_t chunks = N4 / CCBN_CHUNK;
        const uint32_t blocks = chunks / CCBN_KCH;       // 3136 for harness
        if (blocks > 0) {
            uint32_t M = 0, S = 0;
            ccbn_magic_u32(hw4, (uint64_t)N4, &M, &S);
            if (M != 0) {
                ccbn_apply_async<<<blocks, CCBN_CHUNK, 0, stream>>>(
                    (const v4f*)x, ss, (v4f*)d_out, M, S);
                covered4 = blocks * CCBN_KCH * CCBN_CHUNK;
            }
        }
    }

    // Scalar tail for anything not covered (none for harness shapes).
    const uint32_t start = covered4 * 4u;
    if (start < N) {
        const uint32_t rem = N - start;
        ccbn_tail<<<(rem + 255) / 256, 256, 0, stream>>>(
            x, ss, (float*)d_out, start, N, HW);
    }
}